// LSTM_model_84705345012067
// MI455X (gfx1250) — compile-verified
//
#include <hip/hip_runtime.h>
#include <hip/hip_bf16.h>
#include <math.h>

// ---------------------------------------------------------------------------
// LSTM (H=8, I=1, B=4096, T=2048) + FC(8->4) for MI455X (gfx1250, wave32).
//
// Per wave: one tile of 16 batch elements. Recurrent matvec done with
// V_WMMA_F32_16X16X4_F32, oriented as D[m=gate, n=batch] so the weight
// matrix A is loop-invariant and h only needs lane shuffles (no LDS) to
// become the next step's B operand. Input projection + bias folded into the
// matmul as extra K rows (K=8: x * w_ih, K=9: bias, K=10..11: zero).
// Round 3: fix vzero declaration; software-pipelined x loads +
// global_prefetch, fewer cndmasks in activations, critical h regs first.
// ---------------------------------------------------------------------------

typedef __attribute__((ext_vector_type(2))) float v2f;
typedef __attribute__((ext_vector_type(8))) float v8f;

__device__ __forceinline__ float fast_tanh(float x) {
#if __has_builtin(__builtin_amdgcn_tanhf)
  return __builtin_amdgcn_tanhf(x);          // v_tanh_f32 (gfx1250 TRANS op)
#elif __has_builtin(__builtin_amdgcn_tanh_f32)
  return __builtin_amdgcn_tanh_f32(x);
#else
  return tanhf(x);
#endif
}

// sigmoid(x) = 0.5 + 0.5*tanh(x/2)  -> single TRANS op
__device__ __forceinline__ float fast_sigmoid(float x) {
  return fmaf(0.5f, fast_tanh(0.5f * x), 0.5f);
}

__device__ __forceinline__ v8f wmma_f32_k4(v2f a, v2f b, v8f c) {
  return __builtin_amdgcn_wmma_f32_16x16x4_f32(
      /*neg_a=*/false, a, /*neg_b=*/false, b,
      /*c_mod=*/(short)0, c, /*reuse_a=*/false, /*reuse_b=*/false);
}

namespace {
constexpr int kH     = 8;     // hidden size
constexpr int kT     = 2048;  // sequence length
constexpr int kBat   = 4096;  // batch
constexpr int kTileB = 16;    // batch per wave (WMMA N)
}

__global__ __launch_bounds__(32)
void lstm_wmma_kernel(const float* __restrict__ x,      // [B, T, 1]
                      const float* __restrict__ w_ih,   // [32, 1]
                      const float* __restrict__ w_hh,   // [32, 8]
                      const float* __restrict__ b_ih,   // [32]
                      const float* __restrict__ b_hh,   // [32]
                      const float* __restrict__ w_fc,   // [4, 8]
                      const float* __restrict__ b_fc,   // [4]
                      float* __restrict__ out)          // [B, 4]
{
  const int lane  = threadIdx.x;          // 0..31 (wave32)
  const int b0    = blockIdx.x * kTileB;  // first batch row of this tile
  const int m     = lane & 15;            // A: gate row (mod 16); B/D: batch col
  const int khalf = lane >> 4;            // lane-half selects K pairs
  const bool lo   = (lane < 16);

  // --- Constant A operand: [gate 32 x K 12] split into 2 gate tiles x 3 K-chunks.
  // A layout (16x4 f32): reg v, lane L -> element (M = L%16, k = v + 2*(L/16)).
  // K<8: w_hh[gate][K]; K==8: w_ih[gate]; K==9: b_ih+b_hh; K>=10: 0.
  v2f A[2][3];
  #pragma unroll
  for (int n = 0; n < 2; ++n) {
    const int gate = n * 16 + m;
    #pragma unroll
    for (int c = 0; c < 3; ++c) {
      float av[2];
      #pragma unroll
      for (int v = 0; v < 2; ++v) {
        const int K = c * 4 + v + 2 * khalf;
        float val;
        if (K < 8)       val = w_hh[gate * kH + K];
        else if (K == 8) val = w_ih[gate];
        else if (K == 9) val = b_ih[gate] + b_hh[gate];
        else             val = 0.0f;
        av[v] = val;
      }
      A[n][c].x = av[0];
      A[n][c].y = av[1];
    }
  }

  // Loop-invariant per-lane constants:
  const float Bone   = lo ? 1.0f : 0.0f;  // B chunk2 reg1: K=9 ones / K=11 zero
  const float scale1 = lo ? 1.0f : 0.5f;  // tile1: tanh(g) vs tanh(o/2)
  const float addc   = lo ? 0.0f : 0.5f;  // tile1: g vs 0.5+0.5*tanh(o/2)
  const v8f vzero = {};                   // WMMA C=0 (inline constant)

  // h[j], c[j] live at lanes 0..15 (batch = lane), register j = hidden index.
  float h[kH], cst[kH];
  #pragma unroll
  for (int r = 0; r < kH; ++r) { h[r] = 0.0f; cst[r] = 0.0f; }

  // Each lane streams its own batch row of x (lanes 16..31 duplicate 0..15).
  const float* xrow = x + (size_t)(b0 + m) * kT;

  // Software pipeline: xv holds data for the current 4 steps, next iteration's
  // load is issued before the compute so HBM latency overlaps the step chain.
  float4 xv = *reinterpret_cast<const float4*>(xrow);

  for (int t0 = 0; t0 < kT; t0 += 4) {
    // Prefetch a few cachelines ahead within this lane's row (speculative,
    // clamped in-bounds; lowers to global_prefetch_b8).
    {
      int tp = t0 + 128;
      if (tp > kT - 4) tp = kT - 4;
      __builtin_prefetch(xrow + tp, 0, 0);
    }
    // Next iteration's x (clamped so the final iteration re-reads in-bounds).
    const int tn = (t0 + 4 < kT) ? (t0 + 4) : t0;
    const float4 xv_next = *reinterpret_cast<const float4*>(xrow + tn);

    const float xts[4] = {xv.x, xv.y, xv.z, xv.w};

    #pragma unroll
    for (int u = 0; u < 4; ++u) {
      const float xt = xts[u];

      // --- Build B operand (4x16 per chunk): reg r, lane L ->
      //     element (k = c*4 + r + 2*(L/16), n = L%16) = h[b = L%16][k].
      // Lanes 0..15 read h[k] locally; lanes 16..31 need h[k+2] from lane L-16.
      const float s2 = __shfl_xor(h[2], 16, 32);
      const float s3 = __shfl_xor(h[3], 16, 32);
      const float s6 = __shfl_xor(h[6], 16, 32);
      const float s7 = __shfl_xor(h[7], 16, 32);
      v2f B0, B1, B2;
      B0.x = lo ? h[0] : s2;
      B0.y = lo ? h[1] : s3;
      B1.x = lo ? h[4] : s6;
      B1.y = lo ? h[5] : s7;
      B2.x = lo ? xt : 0.0f;   // K=8 row = x_t ; K=10 row = 0
      B2.y = Bone;             // K=9 row = 1.0 ; K=11 row = 0

      // --- gates = [W_hh | w_ih | bias] * [h; x; 1]; two 16x16 gate tiles.
      v8f d0 = wmma_f32_k4(A[0][0], B0, vzero);   // gates 0..15  (i | f)
      v8f d1 = wmma_f32_k4(A[1][0], B0, vzero);   // gates 16..31 (g | o)
      d0 = wmma_f32_k4(A[0][1], B1, d0);
      d1 = wmma_f32_k4(A[1][1], B1, d1);
      d0 = wmma_f32_k4(A[0][2], B2, d0);
      d1 = wmma_f32_k4(A[1][2], B2, d1);

      // --- Activations + state update.
      // D layout: reg r, lanes 0..15 -> (gate_local r, b = L); lanes 16..31 ->
      // (gate_local r+8, b = L-16). Tile0: i at r, f at r+8 (both sigmoid).
      // Tile1: g at r (tanh), o at r+8 (sigmoid).
      // Update h[2],h[3],h[6],h[7] first: the next step's B-operand shuffles
      // depend on them, so the scheduler can hoist those ds_bpermutes.
      const int rorder[kH] = {2, 3, 6, 7, 0, 1, 4, 5};
      #pragma unroll
      for (int ri = 0; ri < kH; ++ri) {
        const int r = rorder[ri];
        const float g0 = d0[r];
        const float g1 = d1[r];

        const float ifv = fast_sigmoid(g0);        // i (lo) / f (hi)
        const float fv  = __shfl_xor(ifv, 16, 32); // f at lanes 0..15

        const float th   = fast_tanh(g1 * scale1);       // tanh(g) / tanh(o/2)
        const float act1 = fmaf(th, scale1, addc);       // g / sigmoid(o)
        const float ov   = __shfl_xor(act1, 16, 32);     // o at lanes 0..15

        cst[r] = fmaf(fv, cst[r], ifv * act1);     // c = f*c + i*g
        h[r]   = ov * fast_tanh(cst[r]);           // h = o*tanh(c)
      }
    }

    xv = xv_next;
  }

  // --- Final FC: out[b, mf] = b_fc[mf] + sum_j h[b][j] * w_fc[mf][j]
  if (lo) {
    float o4[4];
    #pragma unroll
    for (int mf = 0; mf < 4; ++mf) {
      float acc = b_fc[mf];
      #pragma unroll
      for (int j = 0; j < kH; ++j) acc = fmaf(h[j], w_fc[mf * kH + j], acc);
      o4[mf] = acc;
    }
    float4 ov4 = {o4[0], o4[1], o4[2], o4[3]};
    *reinterpret_cast<float4*>(out + (size_t)(b0 + m) * 4) = ov4;
  }
}

extern "C" void kernel_launch(void* const* d_in, const int* in_sizes, int n_in,
                              void* d_out, int out_size, void* d_ws, size_t ws_size,
                              hipStream_t stream) {
  (void)in_sizes; (void)n_in; (void)d_ws; (void)ws_size; (void)out_size;
  const float* x    = (const float*)d_in[0];
  const float* w_ih = (const float*)d_in[1];
  const float* w_hh = (const float*)d_in[2];
  const float* b_ih = (const float*)d_in[3];
  const float* b_hh = (const float*)d_in[4];
  const float* w_fc = (const float*)d_in[5];
  const float* b_fc = (const float*)d_in[6];
  float* out = (float*)d_out;

  dim3 grid(kBat / kTileB);  // 256 waves, one 16-batch tile each
  dim3 block(32);            // one wave32 per block
  lstm_wmma_kernel<<<grid, block, 0, stream>>>(x, w_ih, w_hh, b_ih, b_hh,
                                               w_fc, b_fc, out);
}